// GCN_43593918054547
// MI455X (gfx1250) — compile-verified
//
#include <hip/hip_runtime.h>

typedef float v2f __attribute__((ext_vector_type(2)));
typedef float v8f __attribute__((ext_vector_type(8)));

#define IN_DIM 128
#define HID 64

// ---------------- init: deg=1 (self-loop), zero accumulators ----------------
__global__ void gcn_init(float* __restrict__ deg, float* __restrict__ acc,
                         float* __restrict__ out, int N) {
    int t = blockIdx.x * blockDim.x + threadIdx.x;
    int NF = N * HID;
    if (t < NF) { acc[t] = 0.0f; out[t] = 0.0f; }
    if (t < N)  { deg[t] = 1.0f; }
}

// ---------------- degree count over edges (dst) ----------------
__global__ void gcn_deg(const int* __restrict__ dst, float* __restrict__ deg, int E) {
    int e = blockIdx.x * blockDim.x + threadIdx.x;
    if (e < E) atomicAdd(&deg[dst[e]], 1.0f);
}

// ---------------- dinv = rsqrt(deg)  (deg >= 1 always due to self-loops) ----
__global__ void gcn_dinv(const float* __restrict__ deg, float* __restrict__ dinv, int N) {
    int i = blockIdx.x * blockDim.x + threadIdx.x;
    if (i < N) dinv[i] = rsqrtf(deg[i]);
}

// ---------------- H = A(MxK) @ W(Kx64) via V_WMMA_F32_16X16X4_F32 -----------
// Block = 256 threads = 8 waves. Each wave computes a 16-row x 64-col slab
// as four 16x16 C tiles, accumulating over K in steps of 4 (fp32 WMMA).
// OOB rows: A pointer is clamped to a valid row (unpredicated loads); the
// resulting garbage only lands in C rows that the guarded store never writes.
template <int K>
__global__ void gcn_gemm_wmma(const float* __restrict__ A, const float* __restrict__ W,
                              float* __restrict__ H, int M) {
    const int wave = threadIdx.x >> 5;
    const int lane = threadIdx.x & 31;
    const int rowbase = (blockIdx.x * 8 + wave) * 16;
    if (rowbase >= M) return;                 // wave-uniform: EXEC stays all-1s

    const int r  = lane & 15;                 // M (for A) / N (for B) within tile
    const int kh = (lane >> 4) * 2;           // K sub-offset: 0 or 2
    int arow = rowbase + r;
    if (arow >= M) arow = M - 1;              // clamp: keeps loads in-bounds
    const float* __restrict__ Arow = A + (size_t)arow * K;

    v8f c0 = {}, c1 = {}, c2 = {}, c3 = {};

#pragma unroll 4
    for (int k0 = 0; k0 < K; k0 += 4) {
        // A tile 16x4: lane<16 holds K=k0+0,1 ; lane>=16 holds K=k0+2,3
        // adjacent + 8B aligned -> single global_load_b64
        v2f a = *(const v2f*)(Arow + k0 + kh);

        // B tiles 4x16 (mirrored layout): lane<16 rows k0+0,1 ; lane>=16 rows k0+2,3
        const float* __restrict__ Wk0 = W + (size_t)(k0 + kh) * HID;
        const float* __restrict__ Wk1 = Wk0 + HID;
        v2f b0, b1, b2, b3;
        b0.x = Wk0[r];       b0.y = Wk1[r];
        b1.x = Wk0[16 + r];  b1.y = Wk1[16 + r];
        b2.x = Wk0[32 + r];  b2.y = Wk1[32 + r];
        b3.x = Wk0[48 + r];  b3.y = Wk1[48 + r];

        c0 = __builtin_amdgcn_wmma_f32_16x16x4_f32(false, a, false, b0, (short)0, c0, false, false);
        c1 = __builtin_amdgcn_wmma_f32_16x16x4_f32(false, a, false, b1, (short)0, c1, false, false);
        c2 = __builtin_amdgcn_wmma_f32_16x16x4_f32(false, a, false, b2, (short)0, c2, false, false);
        c3 = __builtin_amdgcn_wmma_f32_16x16x4_f32(false, a, false, b3, (short)0, c3, false, false);
    }

    // C/D layout: VGPR i, lane<16 -> M=i ; lane>=16 -> M=8+i ; N = lane&15
    const int col   = lane & 15;
    const int rhalf = (lane >> 4) * 8;
#pragma unroll
    for (int i = 0; i < 8; ++i) {
        int row = rowbase + rhalf + i;
        if (row < M) {
            float* __restrict__ p = H + (size_t)row * HID + col;
            p[0]  = c0[i];
            p[16] = c1[i];
            p[32] = c2[i];
            p[48] = c3[i];
        }
    }
}

// ---------------- edge scatter: acc[dst] += h[src] * dinv[src]*dinv[dst] ----
// 16 lanes per edge; each lane moves a float4 (row is 256B, fully coalesced)
__global__ void gcn_scatter(const float* __restrict__ h, const int* __restrict__ src,
                            const int* __restrict__ dst, const float* __restrict__ dinv,
                            float* __restrict__ acc, int E) {
    int t = blockIdx.x * blockDim.x + threadIdx.x;
    int e   = t >> 4;
    int sub = t & 15;
    if (e >= E) return;
    int s = src[e];
    int d = dst[e];
    float w = dinv[s] * dinv[d];
    const float4 v = ((const float4*)(h + (size_t)s * HID))[sub];
    float* __restrict__ ap = acc + (size_t)d * HID + sub * 4;
    atomicAdd(ap + 0, v.x * w);
    atomicAdd(ap + 1, v.y * w);
    atomicAdd(ap + 2, v.z * w);
    atomicAdd(ap + 3, v.w * w);
}

// ---------------- finalize: acc += h*dinv^2 (self-loop) + bias, opt. ReLU ---
__global__ void gcn_finalize(float* __restrict__ acc, const float* __restrict__ h,
                             const float* __restrict__ dinv, const float* __restrict__ b,
                             int N, int relu) {
    int t = blockIdx.x * blockDim.x + threadIdx.x;
    if (t >= N * HID) return;
    int i = t >> 6;   // node
    int f = t & 63;   // feature
    float di = dinv[i];
    float v = acc[t] + h[t] * di * di + b[f];
    acc[t] = relu ? fmaxf(v, 0.0f) : v;
}

extern "C" void kernel_launch(void* const* d_in, const int* in_sizes, int n_in,
                              void* d_out, int out_size, void* d_ws, size_t ws_size,
                              hipStream_t stream) {
    const float* x  = (const float*)d_in[0];
    const int*   ei = (const int*)d_in[1];
    const float* W1 = (const float*)d_in[2];
    const float* b1 = (const float*)d_in[3];
    const float* W2 = (const float*)d_in[4];
    const float* b2 = (const float*)d_in[5];
    float* out = (float*)d_out;

    const int N = in_sizes[0] / IN_DIM;   // 100000
    const int E = in_sizes[1] / 2;        // 1000000
    const int NF = N * HID;

    const int* srcp = ei;
    const int* dstp = ei + E;

    float* ws   = (float*)d_ws;
    float* deg  = ws;                               // N
    float* dinv = ws + (size_t)N;                   // N
    float* h    = ws + 2 * (size_t)N;               // N*64 (reused for h1 and h2)
    float* acc  = ws + 2 * (size_t)N + (size_t)NF;  // N*64

    const int B = 256;
    dim3 blk(B);

    // init + normalization
    gcn_init<<<dim3((NF + B - 1) / B), blk, 0, stream>>>(deg, acc, out, N);
    gcn_deg<<<dim3((E + B - 1) / B), blk, 0, stream>>>(dstp, deg, E);
    gcn_dinv<<<dim3((N + B - 1) / B), blk, 0, stream>>>(deg, dinv, N);

    // layer 1
    gcn_gemm_wmma<IN_DIM><<<dim3((N + 127) / 128), blk, 0, stream>>>(x, W1, h, N);
    gcn_scatter<<<dim3(((E * 16) + B - 1) / B), blk, 0, stream>>>(h, srcp, dstp, dinv, acc, E);
    gcn_finalize<<<dim3((NF + B - 1) / B), blk, 0, stream>>>(acc, h, dinv, b1, N, 1);

    // layer 2 (input = acc, output accumulates directly into d_out)
    gcn_gemm_wmma<HID><<<dim3((N + 127) / 128), blk, 0, stream>>>(acc, W2, h, N);
    gcn_scatter<<<dim3(((E * 16) + B - 1) / B), blk, 0, stream>>>(h, srcp, dstp, dinv, out, E);
    gcn_finalize<<<dim3((NF + B - 1) / B), blk, 0, stream>>>(out, h, dinv, b2, N, 0);
}